// LeakyIntegrator_88673894793501
// MI455X (gfx1250) — compile-verified
//
#include <hip/hip_runtime.h>
#include <stdint.h>

#ifndef __has_builtin
#define __has_builtin(x) 0
#endif

#define LI_THRESHOLD 0.5f
#define LI_DECAY     0.01f
#define LI_BLOCK     256                       // 8 wave32s
#define LI_ITEMS     32                        // elements per thread
#define LI_TILE      (LI_BLOCK * LI_ITEMS)     // 8192 floats per block
#define LI_PAD       4                         // pad floats per 32-float group
#define LI_PTILE     (LI_TILE + (LI_TILE / 32) * LI_PAD)   // 9216 floats = 36 KiB

#define NEG_INF (-__builtin_inff())

// 16-byte int vector: exact pointee type of the async b128 builtins
// (clang diagnostic showed param type 'vector_size(4*sizeof(int)) int __device__ *').
typedef __attribute__((__vector_size__(4 * sizeof(int)))) int li_v4i;
typedef __attribute__((address_space(1))) li_v4i* li_gptr;   // global
typedef __attribute__((address_space(3))) li_v4i* li_lptr;   // LDS

// Per-block descriptor for decoupled lookback. Separate aggregate and
// inclusive fields so a reader pairing status==1 with the fields can never
// observe the later inclusive values by mistake.
struct __align__(16) BlockDesc {
  uint32_t status;    // 0 = invalid, 1 = aggregate ready, 2 = inclusive ready
  uint32_t pad0;
  float agg_a, agg_b;
  float inc_a, inc_b;
  uint32_t pad1, pad2;
};  // 32 bytes

// Padded LDS index: +4 floats every 32 floats; keeps every float4 16B-aligned
// (group stride 36 floats = 144 B = 9*16 B) while breaking bank conflicts on
// blocked per-thread reads.
__device__ __forceinline__ unsigned pad_idx(unsigned i) {
  return i + ((i >> 5) << 2);
}

// ---- CDNA5 async global<->LDS data path -----------------------------------
__device__ __forceinline__ void async_g2l_b128(const float* g, float* l) {
#if __has_builtin(__builtin_amdgcn_global_load_async_to_lds_b128)
  __builtin_amdgcn_global_load_async_to_lds_b128(
      (li_gptr)g, (li_lptr)l, 0, 0);
#else
  unsigned loff = (unsigned)(unsigned long long)
      (__attribute__((address_space(3))) void*)l;
  unsigned long long ga = (unsigned long long)g;
  asm volatile("global_load_async_to_lds_b128 %0, %1, off"
               :: "v"(loff), "v"(ga) : "memory");
#endif
}

__device__ __forceinline__ void async_l2g_b128(float* g, const float* l) {
#if __has_builtin(__builtin_amdgcn_global_store_async_from_lds_b128)
  __builtin_amdgcn_global_store_async_from_lds_b128(
      (li_gptr)g, (li_lptr)l, 0, 0);
#else
  unsigned loff = (unsigned)(unsigned long long)
      (const __attribute__((address_space(3))) void*)l;
  unsigned long long ga = (unsigned long long)g;
  asm volatile("global_store_async_from_lds_b128 %0, %1, off"
               :: "v"(ga), "v"(loff) : "memory");
#endif
}

__device__ __forceinline__ void wait_async0() {
#if __has_builtin(__builtin_amdgcn_s_wait_asynccnt)
  __builtin_amdgcn_s_wait_asynccnt(0);
#else
  asm volatile("s_wait_asynccnt 0" ::: "memory");
#endif
}
// ---------------------------------------------------------------------------

__global__ void li_init(uint32_t* w, int n) {
  int i = blockIdx.x * blockDim.x + threadIdx.x;
  if (i < n) w[i] = 0u;
}

__global__ void __launch_bounds__(LI_BLOCK)
li_scan(const float* __restrict__ x, float* __restrict__ out,
        uint32_t* __restrict__ counter, BlockDesc* __restrict__ desc) {
  __shared__ __align__(16) float s_tile[LI_PTILE];
  __shared__ float s_wa[LI_BLOCK / 32], s_wb[LI_BLOCK / 32];
  __shared__ float s_ba, s_bb;        // block exclusive prefix
  __shared__ uint32_t s_bid;

  const int tid  = threadIdx.x;
  const int lane = tid & 31;
  const int wid  = tid >> 5;

  // Atomic ticket => virtual block id in launch order (lookback-safe).
  if (tid == 0)
    s_bid = __hip_atomic_fetch_add(counter, 1u, __ATOMIC_RELAXED,
                                   __HIP_MEMORY_SCOPE_AGENT);
  __syncthreads();
  const uint32_t bid = s_bid;
  const size_t tile_base = (size_t)bid * LI_TILE;

  // ---- stage tile: striped, coalesced async b128 loads into padded LDS ----
#pragma unroll
  for (int j = 0; j < LI_TILE / 4 / LI_BLOCK; ++j) {
    const unsigned e = (unsigned)(j * LI_BLOCK + tid) * 4u;   // float index
    async_g2l_b128(x + tile_base + e, &s_tile[pad_idx(e)]);
  }
  wait_async0();
  __syncthreads();

  // ---- load 32 contiguous elements into registers, fold local transform ---
  const unsigned base = (unsigned)tid * LI_ITEMS;
  float4 r[8];
#pragma unroll
  for (int j = 0; j < 8; ++j)
    r[j] = *(const float4*)&s_tile[pad_idx(base + 4u * j)];

  float aa = 0.0f, ab = NEG_INF;   // thread transform: s -> max(ab, s + aa)
#pragma unroll
  for (int j = 0; j < 8; ++j) {
    const float4 v = r[j];
    const float e[4] = {v.x, v.y, v.z, v.w};
#pragma unroll
    for (int k = 0; k < 4; ++k) {
      const float xv = e[k];
      const bool fire = xv >= LI_THRESHOLD;
      ab = fire ? (ab + xv) : fmaxf(0.0f, ab - LI_DECAY);
      aa = fire ? (aa + xv) : (aa - LI_DECAY);
    }
  }

  // ---- wave32 inclusive scan of thread transforms (Hillis-Steele) ---------
  float sca = aa, scb = ab;
#pragma unroll
  for (int d = 1; d < 32; d <<= 1) {
    const float pa = __shfl_up(sca, d, 32);
    const float pb = __shfl_up(scb, d, 32);
    if (lane >= d) {
      scb = fmaxf(scb, pb + sca);   // uses pre-update sca
      sca = pa + sca;
    }
  }
  if (lane == 31) { s_wa[wid] = sca; s_wb[wid] = scb; }
  __syncthreads();

  // ---- thread 0: wave-exclusive scan, publish, decoupled lookback ---------
  if (tid == 0) {
    float ea = 0.0f, eb = NEG_INF;    // running exclusive across waves
#pragma unroll
    for (int w = 0; w < LI_BLOCK / 32; ++w) {
      const float wa = s_wa[w], wb = s_wb[w];
      s_wa[w] = ea; s_wb[w] = eb;     // replace with wave-exclusive prefix
      eb = fmaxf(wb, eb + wa);
      ea = ea + wa;
    }
    // (ea, eb) is now the block aggregate.
    BlockDesc* me = &desc[bid];
    __hip_atomic_store(&me->agg_a, ea, __ATOMIC_RELAXED, __HIP_MEMORY_SCOPE_AGENT);
    __hip_atomic_store(&me->agg_b, eb, __ATOMIC_RELAXED, __HIP_MEMORY_SCOPE_AGENT);
    if (bid == 0) {
      __hip_atomic_store(&me->inc_a, ea, __ATOMIC_RELAXED, __HIP_MEMORY_SCOPE_AGENT);
      __hip_atomic_store(&me->inc_b, eb, __ATOMIC_RELAXED, __HIP_MEMORY_SCOPE_AGENT);
      __hip_atomic_store(&me->status, 2u, __ATOMIC_RELEASE, __HIP_MEMORY_SCOPE_AGENT);
      s_ba = 0.0f; s_bb = NEG_INF;
    } else {
      __hip_atomic_store(&me->status, 1u, __ATOMIC_RELEASE, __HIP_MEMORY_SCOPE_AGENT);
      float pa = 0.0f, pb = NEG_INF;  // accumulated suffix of the prefix
      int j = (int)bid - 1;
      for (;;) {
        uint32_t st;
        do {
          st = __hip_atomic_load(&desc[j].status, __ATOMIC_ACQUIRE,
                                 __HIP_MEMORY_SCOPE_AGENT);
        } while (st == 0u);
        float ja, jb;
        if (st == 2u) {
          ja = __hip_atomic_load(&desc[j].inc_a, __ATOMIC_RELAXED, __HIP_MEMORY_SCOPE_AGENT);
          jb = __hip_atomic_load(&desc[j].inc_b, __ATOMIC_RELAXED, __HIP_MEMORY_SCOPE_AGENT);
        } else {
          ja = __hip_atomic_load(&desc[j].agg_a, __ATOMIC_RELAXED, __HIP_MEMORY_SCOPE_AGENT);
          jb = __hip_atomic_load(&desc[j].agg_b, __ATOMIC_RELAXED, __HIP_MEMORY_SCOPE_AGENT);
        }
        pb = fmaxf(pb, jb + pa);      // combine(T_j, suffix) — order matters
        pa = ja + pa;
        if (st == 2u) break;
        --j;
      }
      const float ib_ = fmaxf(eb, pb + ea);   // inclusive = combine(prefix, agg)
      const float ia_ = pa + ea;
      __hip_atomic_store(&me->inc_a, ia_, __ATOMIC_RELAXED, __HIP_MEMORY_SCOPE_AGENT);
      __hip_atomic_store(&me->inc_b, ib_, __ATOMIC_RELAXED, __HIP_MEMORY_SCOPE_AGENT);
      __hip_atomic_store(&me->status, 2u, __ATOMIC_RELEASE, __HIP_MEMORY_SCOPE_AGENT);
      s_ba = pa; s_bb = pb;
    }
  }
  __syncthreads();

  // ---- per-thread entering state: E = blockExcl ∘ waveExcl ∘ laneExcl -----
  float la = __shfl_up(sca, 1, 32);
  float lb = __shfl_up(scb, 1, 32);
  if (lane == 0) { la = 0.0f; lb = NEG_INF; }
  const float wea = s_wa[wid], web = s_wb[wid];
  const float t1b = fmaxf(web, s_bb + wea);
  const float t1a = s_ba + wea;
  const float Eb  = fmaxf(lb, t1b + la);
  const float Ea  = t1a + la;
  float s = fmaxf(Eb, Ea);            // E evaluated at initial score 0

  // ---- exact sequential replay of this thread's 32 elements ---------------
#pragma unroll
  for (int j = 0; j < 8; ++j) {
    const float4 v = r[j];
    float4 o;
    s = (v.x >= LI_THRESHOLD) ? (s + v.x) : fmaxf(0.0f, s - LI_DECAY); o.x = s;
    s = (v.y >= LI_THRESHOLD) ? (s + v.y) : fmaxf(0.0f, s - LI_DECAY); o.y = s;
    s = (v.z >= LI_THRESHOLD) ? (s + v.z) : fmaxf(0.0f, s - LI_DECAY); o.z = s;
    s = (v.w >= LI_THRESHOLD) ? (s + v.w) : fmaxf(0.0f, s - LI_DECAY); o.w = s;
    *(float4*)&s_tile[pad_idx(base + 4u * j)] = o;
  }
  __syncthreads();

  // ---- drain tile: striped async b128 stores LDS -> global ----------------
#pragma unroll
  for (int j = 0; j < LI_TILE / 4 / LI_BLOCK; ++j) {
    const unsigned e = (unsigned)(j * LI_BLOCK + tid) * 4u;
    async_l2g_b128(out + tile_base + e, &s_tile[pad_idx(e)]);
  }
  wait_async0();
}

extern "C" void kernel_launch(void* const* d_in, const int* in_sizes, int n_in,
                              void* d_out, int out_size, void* d_ws, size_t ws_size,
                              hipStream_t stream) {
  (void)n_in; (void)out_size; (void)ws_size;
  const float* x = (const float*)d_in[0];
  float* out = (float*)d_out;
  const int n = in_sizes[0];              // 2^25
  const int nblk = n / LI_TILE;           // 4096

  uint32_t* counter = (uint32_t*)d_ws;
  BlockDesc* desc = (BlockDesc*)((char*)d_ws + 16);

  // Zero ticket counter + all descriptors every call (ws is not re-poisoned).
  const int init_words = 4 + nblk * (int)(sizeof(BlockDesc) / 4);
  li_init<<<(init_words + 255) / 256, 256, 0, stream>>>((uint32_t*)d_ws, init_words);
  li_scan<<<nblk, LI_BLOCK, 0, stream>>>(x, out, counter, desc);
}